// QuantumDenseLayer_5549097746630
// MI455X (gfx1250) — compile-verified
//
#include <hip/hip_runtime.h>
#include <math.h>

#define NQ    12
#define DIM   4096
#define DEPTH 4

typedef __attribute__((ext_vector_type(2))) float v2f;
typedef __attribute__((ext_vector_type(8))) float v8f;

// ---------------------------------------------------------------------------
// Setup kernel.
// Blocks 0..11: build 16x16 complex gate matrices (Kron of four Rot gates)
//   ws layout: gates[(l*3+g)*2 + {0:re,1:im}][i*16+j]  (6144 floats)
// Block 12: 48 GF(2) permutation basis ints at ws[6144..6191]:
//   basis[l*12 + k] = composed_perm_l(1 << k)   (perm is linear over GF(2))
// ---------------------------------------------------------------------------
__global__ void qdl_setup_kernel(const float* __restrict__ theta,
                                 float* __restrict__ ws) {
    const int tid = threadIdx.x;
    if (blockIdx.x < 12) {
        const int lg = blockIdx.x;           // layer*3 + group
        const int l  = lg / 3, g = lg % 3;
        const int i  = tid >> 4, j = tid & 15;

        float pr = 1.0f, pi = 0.0f;
        #pragma unroll
        for (int q = 0; q < 4; ++q) {
            const int w = g * 4 + q;                      // wire index
            const float* th = theta + (l * NQ + w) * 3;
            const float phi = th[0], the = th[1], om = th[2];
            const float ch = cosf(0.5f * the), sh = sinf(0.5f * the);
            const float a  = 0.5f * (phi + om), b = 0.5f * (phi - om);
            const float ca = cosf(a), sa = sinf(a);
            const float cb = cosf(b), sb = sinf(b);
            const int bi = (i >> (3 - q)) & 1;
            const int bj = (j >> (3 - q)) & 1;
            // Rot = [[e^{-ia} c, -e^{ib} s], [e^{-ib} s, e^{ia} c]]
            float ur, ui;
            if      (bi == 0 && bj == 0) { ur =  ca * ch; ui = -sa * ch; }
            else if (bi == 0 && bj == 1) { ur = -cb * sh; ui = -sb * sh; }
            else if (bi == 1 && bj == 0) { ur =  cb * sh; ui = -sb * sh; }
            else                         { ur =  ca * ch; ui =  sa * ch; }
            const float nr = pr * ur - pi * ui;
            const float ni = pr * ui + pi * ur;
            pr = nr; pi = ni;
        }
        ws[(lg * 2 + 0) * 256 + tid] = pr;
        ws[(lg * 2 + 1) * 256 + tid] = pi;
    } else if (tid < 48) {
        const int l = tid / 12, k = tid % 12;
        const int r = l % (NQ - 1) + 1;
        int j = 1 << k;
        // composed = p0 ∘ p1 ∘ ... ∘ p11  (apply p11 first)
        for (int w = NQ - 1; w >= 0; --w) {
            const int tg   = (w + r) % NQ;
            const int cbit = (j >> (NQ - 1 - w)) & 1;
            j ^= cbit << (NQ - 1 - tg);
        }
        ((int*)ws)[12 * 2 * 256 + tid] = j;
    }
}

// ---------------------------------------------------------------------------
// Main kernel: one block per batch row. State in LDS, gates in LDS,
// 16x16 complex matmuls on v_wmma_f32_16x16x4_f32.
// ---------------------------------------------------------------------------
__global__ __launch_bounds__(256) void qdl_main_kernel(
        const float* __restrict__ x,
        const float* __restrict__ ws,
        float* __restrict__ out) {
    __shared__ float psiR[DIM];          // 16 KB
    __shared__ float psiI[DIM];          // 16 KB
    __shared__ float gLDS[12 * 2 * 256]; // 24 KB gate stage
    __shared__ int   permB[48];          // GF(2) permutation basis
    __shared__ float red[8 * 12];        // reduce scratch

    const int tid  = threadIdx.x;
    const int b    = blockIdx.x;
    const int lane = tid & 31;
    const int wave = tid >> 5;
    const int half = lane >> 4;   // which 16-lane half of the wave
    const int mn   = lane & 15;   // M (for A/D) or N (for B/D) index

    // stage gate matrices + perm basis into LDS (L2-hot broadcast)
    for (int k = tid; k < 12 * 2 * 256; k += 256) gLDS[k] = ws[k];
    if (tid < 48) permB[tid] = ((const int*)ws)[12 * 2 * 256 + tid];

    float4* psiR4 = reinterpret_cast<float4*>(psiR);
    float4* psiI4 = reinterpret_cast<float4*>(psiI);

    // ---- load row (float4), compute norm (with reference's zero-norm case) ----
    const float4* xr4 = reinterpret_cast<const float4*>(x + (size_t)b * DIM);
    float4 xv[4];
    float ss = 0.0f;
    #pragma unroll
    for (int e = 0; e < 4; ++e) {
        const float4 v = xr4[tid + 256 * e];
        xv[e] = v;
        ss += v.x * v.x + v.y * v.y + v.z * v.z + v.w * v.w;
    }
    for (int o = 16; o > 0; o >>= 1) ss += __shfl_xor(ss, o, 32);
    if (lane == 0) red[wave] = ss;
    __syncthreads();
    float n1sq = 0.0f;
    #pragma unroll
    for (int wv = 0; wv < 8; ++wv) n1sq += red[wv];
    const float x0   = x[(size_t)b * DIM];
    const bool  zero = (sqrtf(n1sq) <= 1e-12f);
    const float x0n  = zero ? 1.0f : x0;
    const float inv  = 1.0f / sqrtf(n1sq - x0 * x0 + x0n * x0n);
    __syncthreads();
    #pragma unroll
    for (int e = 0; e < 4; ++e) {
        float4 v = xv[e];
        v.x = (tid == 0 && e == 0) ? x0n : v.x;
        v.x *= inv; v.y *= inv; v.z *= inv; v.w *= inv;
        psiR4[tid + 256 * e] = v;
        psiI4[tid + 256 * e] = make_float4(0.f, 0.f, 0.f, 0.f);
    }
    __syncthreads();

    // ---- circuit ----
    for (int l = 0; l < DEPTH; ++l) {
        #pragma unroll
        for (int g = 0; g < 3; ++g) {
            const float* Gr = &gLDS[((l * 3 + g) * 2 + 0) * 256];
            const float* Gi = &gLDS[((l * 3 + g) * 2 + 1) * 256];

            // A operands (16x4 f32 layout: VGPR0 holds K=k0/k0+2 by lane half,
            // VGPR1 holds K=k0+1/k0+3). Same gate for both tiles of this wave.
            v2f Ar[4], Ai[4], Ain[4];
            #pragma unroll
            for (int kq = 0; kq < 4; ++kq) {
                const int k0 = kq * 4 + 2 * half;
                const v2f ar = *reinterpret_cast<const v2f*>(&Gr[mn * 16 + k0]);
                const v2f ai = *reinterpret_cast<const v2f*>(&Gi[mn * 16 + k0]);
                Ar[kq]  = ar;
                Ai[kq]  = ai;
                Ain[kq] = (v2f){-ai.x, -ai.y};  // f32 WMMA has no A-neg; VALU negate
            }

            #pragma unroll
            for (int tt = 0; tt < 2; ++tt) {
                const int t = wave + tt * 8;    // tile 0..15, disjoint footprints
                v8f accR = {}; v8f accI = {};
                #pragma unroll
                for (int kq = 0; kq < 4; ++kq) {
                    const int k0 = kq * 4 + 2 * half;
                    v2f Br, Bi;
                    if (g == 0) {
                        const int i0 = k0 * 256 + t * 16 + mn;
                        Br = (v2f){psiR[i0], psiR[i0 + 256]};
                        Bi = (v2f){psiI[i0], psiI[i0 + 256]};
                    } else if (g == 1) {
                        const int i0 = t * 256 + k0 * 16 + mn;
                        Br = (v2f){psiR[i0], psiR[i0 + 16]};
                        Bi = (v2f){psiI[i0], psiI[i0 + 16]};
                    } else {
                        const int i0 = t * 256 + mn * 16 + k0;   // contiguous pair
                        Br = *reinterpret_cast<const v2f*>(&psiR[i0]);
                        Bi = *reinterpret_cast<const v2f*>(&psiI[i0]);
                    }
                    accR = __builtin_amdgcn_wmma_f32_16x16x4_f32(false, Ar[kq],  false, Br, (short)0, accR, false, false);
                    accR = __builtin_amdgcn_wmma_f32_16x16x4_f32(false, Ain[kq], false, Bi, (short)0, accR, false, false);
                    accI = __builtin_amdgcn_wmma_f32_16x16x4_f32(false, Ar[kq],  false, Bi, (short)0, accI, false, false);
                    accI = __builtin_amdgcn_wmma_f32_16x16x4_f32(false, Ai[kq],  false, Br, (short)0, accI, false, false);
                }
                // write back D: element (m = v + 8*half, n = mn); same footprint as reads
                #pragma unroll
                for (int v = 0; v < 8; ++v) {
                    const int m = v + 8 * half;
                    int idx;
                    if      (g == 0) idx = m * 256 + t * 16 + mn;
                    else if (g == 1) idx = t * 256 + m * 16 + mn;
                    else             idx = t * 256 + mn * 16 + m;
                    psiR[idx] = accR[v];
                    psiI[idx] = accI[v];
                }
            }
            __syncthreads();
        }

        // ---- entangler permutation via GF(2)-linear basis ----
        // perm(i) = XOR over set bits k of i of basis[k]; i = tid*16 + e
        const int* bas = &permB[l * 12];
        int pbase = 0;
        #pragma unroll
        for (int bb = 0; bb < 8; ++bb)
            pbase ^= ((tid >> bb) & 1) ? bas[4 + bb] : 0;
        int pe[16];
        #pragma unroll
        for (int e = 0; e < 16; ++e) {
            int v = 0;
            if (e & 1) v ^= bas[0];
            if (e & 2) v ^= bas[1];
            if (e & 4) v ^= bas[2];
            if (e & 8) v ^= bas[3];
            pe[e] = v;
        }
        float rr[16], ri[16];
        #pragma unroll
        for (int e = 0; e < 16; ++e) {
            const int j = pbase ^ pe[e];
            rr[e] = psiR[j];
            ri[e] = psiI[j];
        }
        __syncthreads();
        #pragma unroll
        for (int q = 0; q < 4; ++q) {
            psiR4[tid * 4 + q] = make_float4(rr[4*q], rr[4*q+1], rr[4*q+2], rr[4*q+3]);
            psiI4[tid * 4 + q] = make_float4(ri[4*q], ri[4*q+1], ri[4*q+2], ri[4*q+3]);
        }
        __syncthreads();
    }

    // ---- out[b,k] = sum_i (Re^2 + Im^2) * (1 - 2*bit_{11-k}(i)) ----
    float acc[12];
    #pragma unroll
    for (int k = 0; k < 12; ++k) acc[k] = 0.0f;
    #pragma unroll
    for (int e4 = 0; e4 < 4; ++e4) {
        const float4 a = psiR4[tid * 4 + e4];
        const float4 c = psiI4[tid * 4 + e4];
        const int ib = tid * 16 + e4 * 4;
        const float p0 = a.x * a.x + c.x * c.x;
        const float p1 = a.y * a.y + c.y * c.y;
        const float p2 = a.z * a.z + c.z * c.z;
        const float p3 = a.w * a.w + c.w * c.w;
        const float pv[4] = {p0, p1, p2, p3};
        #pragma unroll
        for (int cc = 0; cc < 4; ++cc) {
            const int i = ib + cc;
            #pragma unroll
            for (int k = 0; k < 12; ++k)
                acc[k] += ((i >> (11 - k)) & 1) ? -pv[cc] : pv[cc];
        }
    }
    #pragma unroll
    for (int k = 0; k < 12; ++k) {
        float v = acc[k];
        for (int o = 16; o > 0; o >>= 1) v += __shfl_xor(v, o, 32);
        if (lane == 0) red[wave * 12 + k] = v;
    }
    __syncthreads();
    if (tid < 12) {
        float s = 0.0f;
        #pragma unroll
        for (int wv = 0; wv < 8; ++wv) s += red[wv * 12 + tid];
        out[(size_t)b * 12 + tid] = s;
    }
}

// ---------------------------------------------------------------------------
extern "C" void kernel_launch(void* const* d_in, const int* in_sizes, int n_in,
                              void* d_out, int out_size, void* d_ws, size_t ws_size,
                              hipStream_t stream) {
    const float* x     = (const float*)d_in[0];   // (B, 4096) f32
    const float* theta = (const float*)d_in[1];   // (4, 12, 3) f32
    float*       out   = (float*)d_out;           // (B, 12) f32
    float*       wsf   = (float*)d_ws;            // 24 KB gates + 192 B perm basis

    const int B = in_sizes[0] / DIM;

    qdl_setup_kernel<<<13, 256, 0, stream>>>(theta, wsf);
    qdl_main_kernel<<<B, 256, 0, stream>>>(x, wsf, out);
}